// MyRGCNConvHistory2_83932250898806
// MI455X (gfx1250) — compile-verified
//
#include <hip/hip_runtime.h>

// RGCN conv + history cache for MI455X (gfx1250, wave32).
// Strategy: per-16-node tile, bucket-sum x[src] by relation (register
// accumulators, scalar relation dispatch), then dense [16x128]@[128x64] per
// relation via V_WMMA_F32_16X16X4_F32 (full f32 = reference precision; the
// problem is gather-bound, so the f32 matrix path costs nothing extra).

#define IN_CH 128
#define HID 64
#define NUM_REL 8
#define NODES_PER_WG 16
#define PAD_ROW 132   // 128 + 4 floats pad -> conflict-free strided A reads

typedef __attribute__((ext_vector_type(2))) float v2f;
typedef __attribute__((ext_vector_type(8))) float v8f;

#define ACC_ADD(R) do { acc[R].x += xv.x; acc[R].y += xv.y; \
                        acc[R].z += xv.z; acc[R].w += xv.w; } while (0)

__global__ __launch_bounds__(256)
void rgcn_hist_kernel(const float* __restrict__ x,      // [N,128]
                      const float* __restrict__ W,      // [8,128,64]
                      const float* __restrict__ hist,   // [N,64]
                      const int*   __restrict__ ptr,    // [N+1]
                      const int*   __restrict__ idx,    // [E] src
                      const int*   __restrict__ etype,  // [E]
                      const int*   __restrict__ hmap,   // [N]
                      const unsigned char* __restrict__ used, // [N] bool
                      const int*   __restrict__ hsize,  // [1]
                      float* __restrict__ out,          // [N,64]
                      int N)
{
    __shared__ float bucket[NUM_REL * NODES_PER_WG * PAD_ROW]; // ~66 KB
    __shared__ float partial[8 * 4 * 32];                      // 4 KB

    const int tid   = threadIdx.x;
    const int wave  = tid >> 5;
    const int lane  = tid & 31;
    const int node0 = blockIdx.x * NODES_PER_WG;
    const int c     = lane * 4;               // 4 floats of the row per lane

    // ---- gather phase: wave w owns local nodes {w, w+8} ----
    for (int half = 0; half < 2; ++half) {
        const int nloc = wave + half * 8;
        const int node = node0 + nloc;

        float4 acc[NUM_REL];
        #pragma unroll
        for (int r = 0; r < NUM_REL; ++r) acc[r] = make_float4(0.f, 0.f, 0.f, 0.f);

        if (node < N) {
            const int e0 = ptr[node];
            const int e1 = ptr[node + 1];
            for (int ebase = e0; ebase < e1; ebase += 32) {
                // coalesced metadata: lane l owns edge ebase+l
                const int my_e  = ebase + lane;
                int my_src = 0, my_rel = NUM_REL;
                if (my_e < e1) {
                    my_src = idx[my_e];
                    my_rel = used[my_src] ? etype[my_e] : NUM_REL;
                }
                // prefetch: 4 lines/lane cover every 128B line of every
                // edge's 512B x-row in this batch (emits global_prefetch_b8)
                #pragma unroll
                for (int p = 0; p < 4; ++p)
                    __builtin_prefetch(x + (size_t)my_src * IN_CH + p * 32, 0, 3);

                const int cnt = min(32, e1 - ebase);
                for (int j = 0; j < cnt; ++j) {
                    // scalar broadcast -> uniform branch + SALU addressing
                    const int src = __builtin_amdgcn_readlane(my_src, j);
                    const int rel = __builtin_amdgcn_readlane(my_rel, j);
                    if (rel < NUM_REL) {
                        const float4 xv =
                            *(const float4*)(x + (size_t)src * IN_CH + c);
                        switch (rel) {
                            case 0: ACC_ADD(0); break;
                            case 1: ACC_ADD(1); break;
                            case 2: ACC_ADD(2); break;
                            case 3: ACC_ADD(3); break;
                            case 4: ACC_ADD(4); break;
                            case 5: ACC_ADD(5); break;
                            case 6: ACC_ADD(6); break;
                            default: ACC_ADD(7); break;
                        }
                    }
                }
            }
        }
        // single store per (rel, node) bucket -- no zero-init, no LDS RMW
        #pragma unroll
        for (int r = 0; r < NUM_REL; ++r)
            *(float4*)&bucket[(r * NODES_PER_WG + nloc) * PAD_ROW + c] = acc[r];
    }
    __syncthreads();

    // ---- compute phase ----
    // waves 0..3: relations 0..3, N-tile = wave
    // waves 4..7: relations 4..7, N-tile = wave-4  (partials merged via LDS)
    const int tileN   = wave & 3;
    const int relBase = (wave >> 2) * 4;
    const int n0      = tileN * 16;

    const int mrow  = lane & 15;           // A row (M) / B,C column (N)
    const int khalf = (lane >> 4) * 2;     // lanes 16..31 hold K+2, K+3

    v8f accd = {0.f, 0.f, 0.f, 0.f, 0.f, 0.f, 0.f, 0.f};

    for (int r = relBase; r < relBase + 4; ++r) {
        const float* Sb = &bucket[(r * NODES_PER_WG + mrow) * PAD_ROW];
        const float* Wb = W + (size_t)r * IN_CH * HID + n0 + mrow;
        #pragma unroll 4
        for (int k = 0; k < IN_CH; k += 4) {
            v2f a, b;
            // A 16x4 f32 fragment (ISA 7.12.2): lane<16 -> K=k,k+1 ; lane>=16 -> K=k+2,k+3
            a.x = Sb[k + khalf + 0];
            a.y = Sb[k + khalf + 1];
            // B 4x16 f32 fragment: row K striped across lanes, half-split like C
            b.x = Wb[(size_t)(k + khalf + 0) * HID];
            b.y = Wb[(size_t)(k + khalf + 1) * HID];
            accd = __builtin_amdgcn_wmma_f32_16x16x4_f32(
                       false, a, false, b, (short)0, accd, false, false);
        }
    }

    // ---- merge partials: waves 4..7 -> LDS, waves 0..3 add ----
    if (wave >= 4) {
        #pragma unroll
        for (int v = 0; v < 8; ++v)
            partial[(v * 4 + tileN) * 32 + lane] = accd[v];
    }
    __syncthreads();

    if (wave < 4) {
        #pragma unroll
        for (int v = 0; v < 8; ++v)
            accd[v] += partial[(v * 4 + tileN) * 32 + lane];

        // C/D layout: VGPR v, lanes 0-15 -> M=v, lanes 16-31 -> M=v+8; N=lane&15
        const int col      = n0 + mrow;
        const int rbase    = (lane >> 4) * 8;
        const bool useHist = hsize[0] > 0;
        #pragma unroll
        for (int v = 0; v < 8; ++v) {
            const int node = node0 + rbase + v;
            if (node < N) {
                const int deg = ptr[node + 1] - ptr[node];
                float val = accd[v] / (float)deg;          // degree norm
                if (useHist && hmap[node] != -1)
                    val = hist[(size_t)node * HID + col];  // history override
                out[(size_t)node * HID + col] = val;
            }
        }
    }
}

extern "C" void kernel_launch(void* const* d_in, const int* in_sizes, int n_in,
                              void* d_out, int out_size, void* d_ws, size_t ws_size,
                              hipStream_t stream) {
    const float* x     = (const float*)d_in[0];
    const float* W     = (const float*)d_in[1];
    const float* hist  = (const float*)d_in[2];
    const int*   ptr   = (const int*)d_in[3];
    const int*   idx   = (const int*)d_in[4];
    const int*   etype = (const int*)d_in[5];
    // d_in[6] = count (recomputed/ignored by reference)
    const int*   hmap  = (const int*)d_in[7];
    const unsigned char* used = (const unsigned char*)d_in[8]; // numpy bool
    const int*   hsize = (const int*)d_in[9];
    float* out = (float*)d_out;

    const int N = in_sizes[0] / IN_CH;
    const int grid = (N + NODES_PER_WG - 1) / NODES_PER_WG;

    rgcn_hist_kernel<<<grid, 256, 0, stream>>>(x, W, hist, ptr, idx, etype,
                                               hmap, used, hsize, out, N);
}